// MaxUnpooling2DMod_75591424410236
// MI455X (gfx1250) — compile-verified
//
#include <hip/hip_runtime.h>

// MaxUnpooling2D scatter-add, shapes static in the reference:
//   in  : [8,128,128,64]  fp32  -> 8,388,608 elems (33.5 MB)
//   idx : [8,128,128,64]  int32 (33.5 MB)
//   out : [8,256,256,64]  fp32  -> 33,554,432 elems (134 MB)
//
// out_flat = b*2^22 + y*16384 + x*64 + c
//   y = idx>>14, x = (idx>>6)&255, c = flat&63, b = flat>>20
// idx in [0, 2^22)  =>  y*16384 + x*64 == idx & ~63
//   => out_flat = ((flat>>20)<<22) | (idx & ~63u) | (flat & 63u)
//
// Memory strategy (MI455X: 23.3 TB/s HBM, 192 MB L2):
//  - zero-fill with RT b128 stores so output lines are L2-resident for atomics
//  - scatter loads (values + indices, read exactly once) are NON-TEMPORAL so
//    the 67 MB of streams don't evict the 134 MB output from the 192 MB L2
//  - duplicate indices must sum -> native global_atomic_add_f32 (relaxed,
//    agent scope), executed at the L2 atomic units

typedef float v4f __attribute__((ext_vector_type(4)));
typedef int   v4i __attribute__((ext_vector_type(4)));

#define IN_ELEMS  (8u * 128u * 128u * 64u)   // 8,388,608  = 2^23
#define OUT_ELEMS (8u * 256u * 256u * 64u)   // 33,554,432 = 2^25

// ---------------------------------------------------------------------------
// Phase 1: zero the output with exact-fit b128 RT stores (keep lines in L2).
// ---------------------------------------------------------------------------
__global__ void unpool_zero_fill(float* __restrict__ out) {
    unsigned int i = blockIdx.x * blockDim.x + threadIdx.x;   // one v4 per thread
    v4f z = {0.f, 0.f, 0.f, 0.f};
    ((v4f* __restrict__)out)[i] = z;                          // global_store_b128
}

// ---------------------------------------------------------------------------
// Phase 2: vectorized scatter-add. 4 consecutive input elements per thread:
// two NT global_load_b128 (values + indices), then 4 global_atomic_add_f32.
// ---------------------------------------------------------------------------
__global__ void unpool_scatter_add(const float* __restrict__ in,
                                   const int*   __restrict__ idx,
                                   float*       __restrict__ out) {
    unsigned int t    = blockIdx.x * blockDim.x + threadIdx.x;
    unsigned int base = t * 4u;                               // flat element base

    // Non-temporal: these streams are touched exactly once; don't let them
    // displace the L2-resident output.
    v4f v  = __builtin_nontemporal_load((const v4f*)in  + t); // b128 th:NT
    v4i id = __builtin_nontemporal_load((const v4i*)idx + t); // b128 th:NT

#pragma unroll
    for (int k = 0; k < 4; ++k) {
        unsigned int flat = base + (unsigned int)k;
        unsigned int o = ((flat >> 20) << 22)                 // batch * 2^22
                       | ((unsigned int)id[k] & ~63u)         // y*16384 + x*64
                       | (flat & 63u);                        // channel
        __hip_atomic_fetch_add(out + o, v[k],
                               __ATOMIC_RELAXED, __HIP_MEMORY_SCOPE_AGENT);
    }
}

// ---------------------------------------------------------------------------
// Launch: same stream => fill ordered before scatter at device scope.
// Exact-fit grids (all sizes are powers of two).
// ---------------------------------------------------------------------------
extern "C" void kernel_launch(void* const* d_in, const int* in_sizes, int n_in,
                              void* d_out, int out_size, void* d_ws, size_t ws_size,
                              hipStream_t stream) {
    const float* in  = (const float*)d_in[0];
    const int*   idx = (const int*)d_in[1];
    float*       out = (float*)d_out;

    const unsigned int threads = 256;

    // Zero fill: OUT_ELEMS/4 v4 stores -> 2^23 threads -> 32768 blocks.
    const unsigned int zero_blocks = (OUT_ELEMS / 4u) / threads;
    unpool_zero_fill<<<zero_blocks, threads, 0, stream>>>(out);

    // Scatter: IN_ELEMS/4 v4 loads -> 2^21 threads -> 8192 blocks.
    const unsigned int scat_blocks = (IN_ELEMS / 4u) / threads;
    unpool_scatter_add<<<scat_blocks, threads, 0, stream>>>(in, idx, out);
}